// GNN_69569880261173
// MI455X (gfx1250) — compile-verified
//
#include <hip/hip_runtime.h>
#include <hip/hip_bf16.h>

#define NN      100000
#define EE      400000
#define GG      5000
#define HH      128
#define NE      (EE + NN)     // edges + self loops
#define NODE_D  52
#define EDGE_D  16

typedef float v2f __attribute__((ext_vector_type(2)));
typedef float v8f __attribute__((ext_vector_type(8)));
typedef unsigned int v4u __attribute__((ext_vector_type(4)));
typedef int v4i __attribute__((ext_vector_type(4)));
typedef int v8i __attribute__((ext_vector_type(8)));

// Tensor Data Mover availability / arity detection
#if defined(__has_builtin)
#  if __has_builtin(__builtin_amdgcn_tensor_load_to_lds)
#    define HAVE_TDM 1
#  else
#    define HAVE_TDM 0
#  endif
#else
#  define HAVE_TDM 0
#endif
#if __has_include(<hip/amd_detail/amd_gfx1250_TDM.h>)
#  define TDM_6ARG 1      // amdgpu-toolchain (therock-10.0 headers): 6-arg builtin
#else
#  define TDM_6ARG 0      // ROCm 7.2: 5-arg builtin
#endif

// ---------------------------------------------------------------------------
// helpers
// ---------------------------------------------------------------------------
__device__ __forceinline__ unsigned enc_f32(float f) {
    unsigned b = __float_as_uint(f);
    return (b & 0x80000000u) ? ~b : (b | 0x80000000u);
}
__device__ __forceinline__ float dec_f32(unsigned u) {
    return (u & 0x80000000u) ? __uint_as_float(u ^ 0x80000000u)
                             : __uint_as_float(~u);
}

__global__ void zero_f32(float* __restrict__ p, int n) {
    int i = blockIdx.x * blockDim.x + threadIdx.x;
    if (i < n) p[i] = 0.0f;
}

#if HAVE_TDM
// Issue one TDM DMA (per wave) loading a [16 x K] f32 tile that starts at
// `gsrc` (row stride K floats) into LDS byte offset `lds_off`, contiguous
// row-major — identical layout to the cooperative copy it replaces.
__device__ __forceinline__ void tdm_load_tile16(const float* gsrc,
                                                unsigned lds_off, int K) {
    unsigned long long ga = (unsigned long long)(uintptr_t)gsrc;
    v4u g0;
    g0.x = 1u;                                            // count=1, user D#
    g0.y = lds_off;                                       // lds_addr
    g0.z = (unsigned)ga;                                  // global_addr[31:0]
    g0.w = (unsigned)((ga >> 32) & 0x1FFFFFFu) | (2u << 30); // addr[56:32]|type=2
    unsigned uk = (unsigned)K;
    v8i g1;
    g1[0] = (int)(2u << 16);            // data_size = 2 (4 bytes); no flags
    g1[1] = (int)(uk << 16);            // tensor_dim0[15:0] (K < 64K)
    g1[2] = (int)(16u << 16);           // tensor_dim0 hi = 0 | tensor_dim1 = 16
    g1[3] = (int)(uk << 16);            // tensor_dim1 hi = 0 | tile_dim0 = K
    g1[4] = 16;                         // tile_dim1 = 16 | tile_dim2 = 0
    g1[5] = (int)uk;                    // tensor_dim0_stride[31:0] = K
    g1[6] = 0;                          // stride hi | tensor_dim1_stride lo
    g1[7] = 0;
    v4i gz = {0, 0, 0, 0};
#if TDM_6ARG
    v8i gz8 = {0, 0, 0, 0, 0, 0, 0, 0};
    __builtin_amdgcn_tensor_load_to_lds(g0, g1, gz, gz, gz8, 0);
#else
    __builtin_amdgcn_tensor_load_to_lds(g0, g1, gz, gz, 0);
#endif
}
#endif

// ---------------------------------------------------------------------------
// node / edge feature encoders
// ---------------------------------------------------------------------------
__global__ void enc_nodes(const float* __restrict__ x,
                          const float* __restrict__ atom_emb,
                          const float* __restrict__ Wn,
                          const float* __restrict__ bn,
                          const float* __restrict__ bool_emb,
                          float* __restrict__ h0) {
    int n = blockIdx.x * blockDim.x + threadIdx.x;
    if (n >= NN) return;
    const float* xp = x + (long)n * 14;
    float* hp = h0 + (long)n * NODE_D;
    int ai = (int)xp[0];
    #pragma unroll
    for (int j = 0; j < 16; ++j) hp[j] = atom_emb[ai * 16 + j];
    for (int j = 0; j < 30; ++j) {
        float s = bn[j];
        #pragma unroll
        for (int i = 0; i < 10; ++i) s += xp[1 + i] * Wn[i * 30 + j];
        hp[16 + j] = s;
    }
    #pragma unroll
    for (int t = 0; t < 3; ++t) {
        int b = (int)xp[11 + t];
        hp[46 + 2 * t]     = bool_emb[2 * b];
        hp[46 + 2 * t + 1] = bool_emb[2 * b + 1];
    }
}

__global__ void enc_edges(const float* __restrict__ ea,
                          const float* __restrict__ bond_emb,
                          const float* __restrict__ We,
                          const float* __restrict__ be,
                          const float* __restrict__ bool_emb,
                          float* __restrict__ efull) {
    int i = blockIdx.x * blockDim.x + threadIdx.x;
    if (i >= EE) return;
    const float* ap = ea + (long)i * 5;
    float* ep = efull + (long)i * EDGE_D;
    int bi = (int)ap[0];
    #pragma unroll
    for (int j = 0; j < 8; ++j) ep[j] = bond_emb[bi * 8 + j];
    float c = ap[1];
    ep[8] = c * We[0] + be[0];
    ep[9] = c * We[1] + be[1];
    #pragma unroll
    for (int t = 0; t < 3; ++t) {
        int b = (int)ap[2 + t];
        ep[10 + 2 * t] = bool_emb[2 * b];
        ep[11 + 2 * t] = bool_emb[2 * b + 1];
    }
}

// self-loop attr: mean of in-edge attrs per node
__global__ void scatter_edge_sum(const float* __restrict__ efull,
                                 const int* __restrict__ dst,
                                 float* __restrict__ cnt,
                                 float* __restrict__ sume) {
    int idx = blockIdx.x * blockDim.x + threadIdx.x;   // E*16 threads
    if (idx >= EE * EDGE_D) return;
    int e = idx >> 4, k = idx & 15;
    int d = dst[e];
    if (k == 0) atomicAdd(&cnt[d], 1.0f);
    atomicAdd(&sume[d * EDGE_D + k], efull[(long)e * EDGE_D + k]);
}

__global__ void write_loop_attr(const float* __restrict__ cnt,
                                const float* __restrict__ sume,
                                float* __restrict__ efull) {
    int idx = blockIdx.x * blockDim.x + threadIdx.x;   // N*16 threads
    if (idx >= NN * EDGE_D) return;
    int n = idx >> 4, k = idx & 15;
    float c = cnt[n];
    efull[(long)(EE + n) * EDGE_D + k] = sume[n * EDGE_D + k] / fmaxf(c, 1.0f);
}

// ---------------------------------------------------------------------------
// WMMA f32 GEMM:  Y[M x 128] = A[M x K] @ W[K x 128] (+ bias)
// block = 256 threads (8 waves); wave w owns columns 16w..16w+15,
// block owns rows 16*blockIdx.x .. +15.  K % 4 == 0, M % 16 == 0.
// A tile staged to LDS by the Tensor Data Mover (one DMA per block).
// ---------------------------------------------------------------------------
__global__ __launch_bounds__(256) void gemm_wmma_f32(
        const float* __restrict__ A, const float* __restrict__ W,
        const float* __restrict__ bias, float* __restrict__ Y,
        int M, int K) {
    __shared__ float As[16 * 128];
    const int tid = threadIdx.x;
    const int m0  = blockIdx.x * 16;

#if HAVE_TDM
    if (tid < 32) {
        tdm_load_tile16(A + (long)m0 * K, (unsigned)(uintptr_t)&As[0], K);
        __builtin_amdgcn_s_wait_tensorcnt(0);
    }
    __syncthreads();
#else
    for (int idx = tid; idx < 16 * K; idx += 256) {
        int r = idx / K, c = idx - r * K;
        As[idx] = A[(long)(m0 + r) * K + c];
    }
    __syncthreads();
#endif

    const int lane = tid & 31;
    const int wave = tid >> 5;
    const int half = lane >> 4;       // 0: lanes 0-15, 1: lanes 16-31
    const int mr   = lane & 15;       // A row within tile
    const int nl   = lane & 15;       // B/D column within tile
    const int n0   = wave * 16;
    const int col  = n0 + nl;

    v8f c = {};
    for (int k = 0; k < K; k += 4) {
        v2f a, b;
        a.x = As[mr * K + k + 2 * half];
        a.y = As[mr * K + k + 2 * half + 1];
        const float* wp = W + (long)(k + 2 * half) * HH + col;
        b.x = wp[0];
        b.y = wp[HH];
        if (k + 4 < K) __builtin_prefetch(W + (long)(k + 4) * HH + col, 0, 1);
        c = __builtin_amdgcn_wmma_f32_16x16x4_f32(
                false, a, false, b, (short)0, c, false, false);
    }

    float bb = bias ? bias[col] : 0.0f;
    #pragma unroll
    for (int r = 0; r < 8; ++r) {
        int row = m0 + r + 8 * half;
        Y[(long)row * HH + col] = c[r] + bb;
    }
}

// ---------------------------------------------------------------------------
// attention: one wave per edge computes a = leaky_relu(xl[s]+xr[d]+elin)@att
// ---------------------------------------------------------------------------
__global__ void edge_score(const float* __restrict__ xl,
                           const float* __restrict__ xr,
                           const float* __restrict__ elin,
                           const int* __restrict__ src,
                           const int* __restrict__ dst,
                           const float* __restrict__ att,
                           float* __restrict__ a_out,
                           unsigned* __restrict__ amax) {
    int gid  = blockIdx.x * blockDim.x + threadIdx.x;
    int edge = gid >> 5;
    int lane = gid & 31;
    if (edge >= NE) return;
    int s, d;
    if (edge < EE) { s = src[edge]; d = dst[edge]; }
    else           { s = d = edge - EE; }
    float acc = 0.0f;
    #pragma unroll
    for (int j = 0; j < 4; ++j) {
        int hh = lane + 32 * j;
        float m = xl[(long)s * HH + hh] + xr[(long)d * HH + hh]
                + elin[(long)edge * HH + hh];
        float lr = m > 0.0f ? m : 0.2f * m;
        acc += lr * att[hh];
    }
    #pragma unroll
    for (int off = 16; off > 0; off >>= 1) acc += __shfl_xor(acc, off, 32);
    if (lane == 0) {
        a_out[edge] = acc;
        atomicMax(&amax[d], enc_f32(acc));
    }
}

__global__ void edge_exp(const float* __restrict__ a,
                         const int* __restrict__ dst,
                         const unsigned* __restrict__ amax,
                         float* __restrict__ ex,
                         float* __restrict__ denom) {
    int i = blockIdx.x * blockDim.x + threadIdx.x;
    if (i >= NE) return;
    int d = (i < EE) ? dst[i] : (i - EE);
    float e = __expf(a[i] - dec_f32(amax[d]));
    ex[i] = e;
    atomicAdd(&denom[d], e);
}

__global__ void aggregate(const float* __restrict__ ex,
                          const float* __restrict__ denom,
                          const float* __restrict__ xl,
                          const int* __restrict__ src,
                          const int* __restrict__ dst,
                          float* __restrict__ hacc) {
    long idx = (long)blockIdx.x * blockDim.x + threadIdx.x;  // NE*128
    if (idx >= (long)NE * HH) return;
    int edge = (int)(idx >> 7);
    int hh   = (int)(idx & 127);
    int s, d;
    if (edge < EE) { s = src[edge]; d = dst[edge]; }
    else           { s = d = edge - EE; }
    float alpha = ex[edge] / denom[d];
    atomicAdd(&hacc[(long)d * HH + hh], alpha * xl[(long)s * HH + hh]);
}

__global__ void finalize_relu(const float* __restrict__ hacc,
                              const float* __restrict__ bias,
                              float* __restrict__ hcur) {
    int i = blockIdx.x * blockDim.x + threadIdx.x;
    if (i >= NN * HH) return;
    hcur[i] = fmaxf(hacc[i] + bias[i & 127], 0.0f);
}

// global_max_pool: batch = repeat(arange(G), 20)
__global__ void pool_max(const float* __restrict__ hcur, float* __restrict__ out) {
    int idx = blockIdx.x * blockDim.x + threadIdx.x;
    if (idx >= GG * HH) return;
    int g = idx >> 7, hh = idx & 127;
    float best = hcur[(long)(g * 20) * HH + hh];
    #pragma unroll
    for (int j = 1; j < 20; ++j)
        best = fmaxf(best, hcur[(long)(g * 20 + j) * HH + hh]);
    out[idx] = best;
}

// ---------------------------------------------------------------------------
// host driver
// ---------------------------------------------------------------------------
static inline int cdiv(long a, int b) { return (int)((a + b - 1) / b); }

extern "C" void kernel_launch(void* const* d_in, const int* in_sizes, int n_in,
                              void* d_out, int out_size, void* d_ws, size_t ws_size,
                              hipStream_t stream) {
    const float* x        = (const float*)d_in[0];
    const int*   ei       = (const int*)  d_in[1];   // [2, E]
    const float* ea       = (const float*)d_in[2];
    const float* atom_emb = (const float*)d_in[4];
    const float* bond_emb = (const float*)d_in[5];
    const float* bool_emb = (const float*)d_in[6];
    const float* Wn  = (const float*)d_in[7];
    const float* bn  = (const float*)d_in[8];
    const float* We  = (const float*)d_in[9];
    const float* be  = (const float*)d_in[10];
    const float* Wl1 = (const float*)d_in[11];
    const float* bl1 = (const float*)d_in[12];
    const float* Wr1 = (const float*)d_in[13];
    const float* br1 = (const float*)d_in[14];
    const float* Wedge1 = (const float*)d_in[15];
    const float* att1   = (const float*)d_in[16];
    const float* bias1  = (const float*)d_in[17];
    const float* Wl2 = (const float*)d_in[18];
    const float* bl2 = (const float*)d_in[19];
    const float* Wr2 = (const float*)d_in[20];
    const float* br2 = (const float*)d_in[21];
    const float* Wedge2 = (const float*)d_in[22];
    const float* att2   = (const float*)d_in[23];
    const float* bias2  = (const float*)d_in[24];

    const int* srcv = ei;
    const int* dstv = ei + EE;

    // workspace carve (256B aligned)
    char* ws = (char*)d_ws;
    size_t off = 0;
    auto carve = [&](size_t bytes) {
        char* p = ws + off;
        off = (off + bytes + 255) & ~(size_t)255;
        return p;
    };
    float*    h0    = (float*)   carve((size_t)NN * NODE_D * 4);
    float*    hcur  = (float*)   carve((size_t)NN * HH * 4);
    float*    xl    = (float*)   carve((size_t)NN * HH * 4);
    float*    xr    = (float*)   carve((size_t)NN * HH * 4);
    float*    hacc  = (float*)   carve((size_t)NN * HH * 4);
    float*    efull = (float*)   carve((size_t)NE * EDGE_D * 4);
    float*    elin  = (float*)   carve((size_t)NE * HH * 4);
    float*    a_e   = (float*)   carve((size_t)NE * 4);
    float*    ex_e  = (float*)   carve((size_t)NE * 4);
    unsigned* amax  = (unsigned*)carve((size_t)NN * 4);
    float*    denom = (float*)   carve((size_t)NN * 4);
    float*    cnt   = (float*)   carve((size_t)NN * 4);
    float*    sume  = (float*)   carve((size_t)NN * EDGE_D * 4);

    const int T = 256;

    // --- encode nodes + edges ---
    enc_nodes<<<cdiv(NN, T), T, 0, stream>>>(x, atom_emb, Wn, bn, bool_emb, h0);
    enc_edges<<<cdiv(EE, T), T, 0, stream>>>(ea, bond_emb, We, be, bool_emb, efull);

    // --- self-loop attrs (mean of in-edges) ---
    zero_f32<<<cdiv(NN, T), T, 0, stream>>>(cnt, NN);
    zero_f32<<<cdiv((long)NN * EDGE_D, T), T, 0, stream>>>(sume, NN * EDGE_D);
    scatter_edge_sum<<<cdiv((long)EE * EDGE_D, T), T, 0, stream>>>(efull, dstv, cnt, sume);
    write_loop_attr<<<cdiv((long)NN * EDGE_D, T), T, 0, stream>>>(cnt, sume, efull);

    // one GATv2 layer
    auto layer = [&](const float* hin, int K,
                     const float* Wl, const float* bl,
                     const float* Wr, const float* br,
                     const float* att, const float* bias) {
        gemm_wmma_f32<<<NN / 16, 256, 0, stream>>>(hin, Wl, bl, xl, NN, K);
        gemm_wmma_f32<<<NN / 16, 256, 0, stream>>>(hin, Wr, br, xr, NN, K);
        zero_f32<<<cdiv(NN, T), T, 0, stream>>>((float*)amax, NN);
        zero_f32<<<cdiv(NN, T), T, 0, stream>>>(denom, NN);
        zero_f32<<<cdiv((long)NN * HH, T), T, 0, stream>>>(hacc, NN * HH);
        edge_score<<<cdiv((long)NE * 32, T), T, 0, stream>>>(xl, xr, elin, srcv, dstv,
                                                             att, a_e, amax);
        edge_exp<<<cdiv(NE, T), T, 0, stream>>>(a_e, dstv, amax, ex_e, denom);
        aggregate<<<cdiv((long)NE * HH, T), T, 0, stream>>>(ex_e, denom, xl, srcv,
                                                            dstv, hacc);
        finalize_relu<<<cdiv((long)NN * HH, T), T, 0, stream>>>(hacc, bias, hcur);
    };

    // layer 1 (uses Wedge1)
    gemm_wmma_f32<<<NE / 16, 256, 0, stream>>>(efull, Wedge1, nullptr, elin, NE, EDGE_D);
    layer(h0, NODE_D, Wl1, bl1, Wr1, br1, att1, bias1);

    // layers 2 & 3 (shared weights, Wedge2)
    gemm_wmma_f32<<<NE / 16, 256, 0, stream>>>(efull, Wedge2, nullptr, elin, NE, EDGE_D);
    layer(hcur, HH, Wl2, bl2, Wr2, br2, att2, bias2);
    layer(hcur, HH, Wl2, bl2, Wr2, br2, att2, bias2);

    // readout
    pool_max<<<cdiv((long)GG * HH, T), T, 0, stream>>>(hcur, (float*)d_out);
}